// ActMLPNetwork_63264868270566
// MI455X (gfx1250) — compile-verified
//
#include <hip/hip_runtime.h>

#define IN_DIM   32
#define OUT_DIM  16
#define NUM_BINS 1100
#define EMBED    8
#define HEADS    4
#define BATCH    1024
#define MN_V     (-1.0f)
#define MX_V     (300.0f)

typedef __attribute__((ext_vector_type(2))) float v2f;
typedef __attribute__((ext_vector_type(8))) float v8f;

// ---------------------------------------------------------------------------
// Kernel 1: uniform-bin searchsorted + embedding gather + fused QKV projection
// q pre-scaled by 1/sqrt(head_dim). Output layout: [f*4+h][b] as float2 (d=2).
// ---------------------------------------------------------------------------
__global__ __launch_bounds__(256) void qkv_kernel(
    const float* __restrict__ x,        // (1024, 32)
    const float* __restrict__ emb,      // (32, 1100, 8)
    const float* __restrict__ ipw,      // (24, 8)
    const float* __restrict__ ipb,      // (24,)
    float2* __restrict__ qbuf,          // (128, 1024) float2
    float2* __restrict__ kbuf,
    float2* __restrict__ vbuf)
{
    __shared__ float sW[192];
    __shared__ float sB[24];
    int t = threadIdx.x;
    if (t < 192) sW[t] = ipw[t];
    if (t >= 192 && t < 216) sB[t - 192] = ipb[t - 192];
    __syncthreads();

    int gid = blockIdx.x * 256 + t;     // 0..32767
    int b = gid >> 5;
    int f = gid & 31;

    // searchsorted(bins, clip(x), side='left') on uniform bins
    float v = x[b * IN_DIM + f];
    v = fminf(fmaxf(v, MN_V), MX_V);
    const float step = (MX_V - MN_V) / (float)(NUM_BINS - 1);
    int i = (int)ceilf((v - MN_V) / step);
    i = min(max(i, 0), NUM_BINS - 1);
    while (i > 0 && (MN_V + (float)(i - 1) * step) >= v) --i;
    while (i < NUM_BINS - 1 && (MN_V + (float)i * step) < v) ++i;

    const float4* ep = (const float4*)(emb + ((size_t)f * NUM_BINS + (size_t)i) * EMBED);
    float4 e0 = ep[0], e1 = ep[1];
    float e[8] = {e0.x, e0.y, e0.z, e0.w, e1.x, e1.y, e1.z, e1.w};

    float out[24];
#pragma unroll
    for (int j = 0; j < 24; ++j) {
        float s = sB[j];
#pragma unroll
        for (int ii = 0; ii < 8; ++ii) s += e[ii] * sW[j * 8 + ii];
        out[j] = s;
    }

    const float rs2 = 0.70710678118654752440f;   // 1/sqrt(2)
#pragma unroll
    for (int h = 0; h < HEADS; ++h) {
        int fh = f * HEADS + h;
        qbuf[(size_t)fh * BATCH + b] = make_float2(out[h * 2] * rs2, out[h * 2 + 1] * rs2);
        kbuf[(size_t)fh * BATCH + b] = make_float2(out[8 + h * 2], out[8 + h * 2 + 1]);
        vbuf[(size_t)fh * BATCH + b] = make_float2(out[16 + h * 2], out[16 + h * 2 + 1]);
    }
}

// ---------------------------------------------------------------------------
// Kernel 2: flash attention per (f,h). One wave owns 16 batch rows (columns of
// the S^T tile). Per 16-c chunk: one v_wmma_f32_16x16x4 computes S^T, lane-
// local online softmax (rows of a column live in the lane pair n / n+16),
// AV accumulated from LDS-resident V. Pair merged once at the end.
//
// Branchless hot loop: the B operand (Q^T) has its K-rows 2..3 exactly zero
// (hi-lane qx/qy stay 0), so the A operand's hi lanes may hold arbitrary
// *finite* K values -- their products are multiplied by 0.0. All 32 lanes
// therefore load sK[cb+n] unconditionally (LDS same-address broadcast).
// ---------------------------------------------------------------------------
__global__ __launch_bounds__(512) void attn_kernel(
    const float2* __restrict__ qbuf,
    const float2* __restrict__ kbuf,
    const float2* __restrict__ vbuf,
    float* __restrict__ obuf)           // (1024, 32, 8)
{
    __shared__ float2 sK[BATCH];
    __shared__ float2 sV[BATCH];

    int fh      = blockIdx.x >> 2;
    int quarter = blockIdx.x & 3;
    int lane    = threadIdx.x & 31;
    int wave    = threadIdx.x >> 5;
    int n       = lane & 15;
    bool lo     = lane < 16;

    const float2* kp = kbuf + (size_t)fh * BATCH;
    const float2* vp = vbuf + (size_t)fh * BATCH;
    for (int idx = threadIdx.x; idx < BATCH; idx += 512) {
        sK[idx] = kp[idx];
        sV[idx] = vp[idx];
    }
    __syncthreads();

    int b = quarter * 256 + wave * 16 + n;

    float qx = 0.f, qy = 0.f;
    if (lo) { float2 q = qbuf[(size_t)fh * BATCH + b]; qx = q.x; qy = q.y; }
    v2f bm; bm[0] = qx; bm[1] = qy;      // B = Q^T (K-rows 2,3 are zero lanes)

    const float NEG_INF = -__builtin_inff();
    float m = NEG_INF, l = 0.f, ox = 0.f, oy = 0.f;
    int half8 = lo ? 0 : 8;

#pragma unroll 2
    for (int cb = 0; cb < BATCH; cb += 16) {
        float2 kk = sK[cb + n];              // unconditional: see note above
        v2f am; am[0] = kk.x; am[1] = kk.y;  // A = K chunk
        v8f acc = {};
        acc = __builtin_amdgcn_wmma_f32_16x16x4_f32(false, am, false, bm,
                                                    (short)0, acc, false, false);
        // lane holds S^T[c = cb + r + half8][b] in acc[r], r = 0..7
        float cm = acc[0];
#pragma unroll
        for (int r = 1; r < 8; ++r) cm = fmaxf(cm, acc[r]);
        float mnew  = fmaxf(m, cm);
        float alpha = __expf(m - mnew);
        l *= alpha; ox *= alpha; oy *= alpha; m = mnew;
        int cbase = cb + half8;
#pragma unroll
        for (int r = 0; r < 8; ++r) {
            float p = __expf(acc[r] - mnew);
            float2 vv = sV[cbase + r];
            l += p; ox += p * vv.x; oy += p * vv.y;
        }
    }

    // merge online-softmax state of the lane pair (n, n+16)
    float m2  = __shfl_xor(m, 16, 32);
    float l2  = __shfl_xor(l, 16, 32);
    float ox2 = __shfl_xor(ox, 16, 32);
    float oy2 = __shfl_xor(oy, 16, 32);
    float mm = fmaxf(m, m2);
    float a1 = __expf(m - mm), a2 = __expf(m2 - mm);
    float L  = l * a1 + l2 * a2;
    float OX = (ox * a1 + ox2 * a2) / L;
    float OY = (oy * a1 + oy2 * a2) / L;

    if (lo) {
        int f = fh >> 2, h = fh & 3;
        float* op = obuf + ((size_t)b * IN_DIM + f) * EMBED + h * 2;
        op[0] = OX;
        op[1] = OY;
    }
}

// ---------------------------------------------------------------------------
// Kernel 3: out_proj (8x8, VALU) -> LDS, then logits GEMM 16x16x256 as 64
// chained v_wmma_f32_16x16x4, bias + row softmax via small LDS tile.
// One wave per 16 batch rows.
// ---------------------------------------------------------------------------
__global__ __launch_bounds__(32) void head_kernel(
    const float* __restrict__ obuf,     // (1024, 32, 8)
    const float* __restrict__ opw,      // (8, 8)
    const float* __restrict__ opb,      // (8,)
    const float* __restrict__ linw,     // (16, 256)
    const float* __restrict__ linb,     // (16,)
    float* __restrict__ out)            // (1024, 16)
{
    __shared__ float sOO[16 * 256];     // post-out_proj activations [b_local][k]
    __shared__ float sW[16 * 256];      // lin_w [j][k]
    __shared__ float sPW[64];
    __shared__ float sPB[8];
    __shared__ float sSM[16 * 16];

    int lane  = threadIdx.x;
    int bbase = blockIdx.x * 16;

    for (int t = lane; t < 4096; t += 32) sW[t] = linw[t];
    for (int t = lane; t < 64; t += 32)   sPW[t] = opw[t];
    if (lane < 8) sPB[lane] = opb[lane];
    __syncthreads();

    for (int t = lane; t < 4096; t += 32) {
        int bl = t >> 8, oc = t & 255;
        int f = oc >> 3, j = oc & 7;
        const float* orow = obuf + ((size_t)(bbase + bl) * IN_DIM + f) * EMBED;
        float s = sPB[j];
#pragma unroll
        for (int ii = 0; ii < 8; ++ii) s += orow[ii] * sPW[j * 8 + ii];
        sOO[t] = s;
    }
    __syncthreads();

    int n = lane & 15;
    int koffadd = (lane >> 4) << 1;     // lanes >=16 supply K-dims 2,3
    v8f acc = {};
    for (int k0 = 0; k0 < 256; k0 += 4) {
        int k = k0 + koffadd;
        v2f am;  am[0]  = sOO[n * 256 + k]; am[1]  = sOO[n * 256 + k + 1];
        v2f bmv; bmv[0] = sW[n * 256 + k];  bmv[1] = sW[n * 256 + k + 1];
        acc = __builtin_amdgcn_wmma_f32_16x16x4_f32(false, am, false, bmv,
                                                    (short)0, acc, false, false);
    }

    float bias = linb[n];
    int rowadd = (lane >> 4) << 3;
#pragma unroll
    for (int r = 0; r < 8; ++r)
        sSM[(r + rowadd) * 16 + n] = acc[r] + bias;
    __syncthreads();

    if (lane < 16) {
        const float* row = sSM + lane * 16;
        float mx = row[0];
#pragma unroll
        for (int j = 1; j < 16; ++j) mx = fmaxf(mx, row[j]);
        float ev[16];
        float sum = 0.f;
#pragma unroll
        for (int j = 0; j < 16; ++j) { ev[j] = __expf(row[j] - mx); sum += ev[j]; }
        float inv = 1.f / sum;
        float* op = out + (size_t)(bbase + lane) * OUT_DIM;
#pragma unroll
        for (int j = 0; j < 16; ++j) op[j] = ev[j] * inv;
    }
}

// ---------------------------------------------------------------------------
extern "C" void kernel_launch(void* const* d_in, const int* in_sizes, int n_in,
                              void* d_out, int out_size, void* d_ws, size_t ws_size,
                              hipStream_t stream)
{
    const float* x    = (const float*)d_in[0];
    const float* emb  = (const float*)d_in[1];
    const float* ipw  = (const float*)d_in[2];
    const float* ipb  = (const float*)d_in[3];
    const float* opw  = (const float*)d_in[4];
    const float* opb  = (const float*)d_in[5];
    const float* linw = (const float*)d_in[6];
    const float* linb = (const float*)d_in[7];
    float* outp = (float*)d_out;

    char* ws = (char*)d_ws;
    float2* qbuf = (float2*)(ws + (size_t)0);
    float2* kbuf = (float2*)(ws + (size_t)(1u << 20));
    float2* vbuf = (float2*)(ws + (size_t)(2u << 20));
    float*  obuf = (float*) (ws + (size_t)(3u << 20));

    hipLaunchKernelGGL(qkv_kernel,  dim3(128), dim3(256), 0, stream,
                       x, emb, ipw, ipb, qbuf, kbuf, vbuf);
    hipLaunchKernelGGL(attn_kernel, dim3(512), dim3(512), 0, stream,
                       qbuf, kbuf, vbuf, obuf);
    hipLaunchKernelGGL(head_kernel, dim3(64),  dim3(32),  0, stream,
                       obuf, opw, opb, linw, linb, outp);
}